// DifferentiableBundleAdjustment_67010079752540
// MI455X (gfx1250) — compile-verified
//
#include <hip/hip_runtime.h>

// Problem constants (fixed by the reference).
#define BQ   4096
#define SQ   512
#define PQ   32       // dba_params channels
#define SDQ  15       // state dim
#define NB   32       // batches per block == one wave32 (one batch per lane)
#define TC   8        // timesteps per chunk
#define NCHUNK (SQ / TC)     // 64
#define OSTRIDE 124          // dwords per thread region in out tile (8*15=120 + 4 pad; 496B, 16B aligned)
#define OBUFSZ  (NB * OSTRIDE)

typedef float f4 __attribute__((ext_vector_type(4)));

// Low 32 bits of a flat pointer to a __shared__ object == LDS byte offset
// (flat LDS aperture keeps the offset in addr[31:0] per CDNA5 ISA §10.2).
__device__ __forceinline__ unsigned lds_off(const void* p) {
    return (unsigned)(unsigned long long)p;
}

struct State { float s0, s1, s2, q0, q1, q2, q3; };

// 16 non-temporal b128 loads -> 64 VGPRs; tracked on LOADcnt, so the compiler
// software-pipelines them against compute and the ASYNCcnt store DMA.
__device__ __forceinline__ void load_chunk(const float* __restrict__ ibase, int t0,
                                           f4 (&buf)[16]) {
    #pragma unroll
    for (int t = 0; t < TC; ++t) {
        const f4* r = (const f4*)(ibase + (size_t)(t0 + t) * PQ);
        buf[2 * t]     = __builtin_nontemporal_load(r);
        buf[2 * t + 1] = __builtin_nontemporal_load(r + 1);
    }
}

// Emit current state (reference emits BEFORE applying delta[t]), then update.
__device__ __forceinline__ void compute_emit(State& st, const f4 (&buf)[16],
                                             float* __restrict__ orow0) {
    #pragma unroll
    for (int t = 0; t < TC; ++t) {
        float* orow = orow0 + t * SDQ;               // packed 15-dword rows
        orow[0] = st.s0; orow[1] = st.s1; orow[2] = st.s2; orow[3] = st.q0;
        orow[4] = st.q1; orow[5] = st.q2; orow[6] = st.q3;
        f4 dA = buf[2 * t], dB = buf[2 * t + 1];
        st.s0 += 0.1f * dA.x;  st.s1 += 0.1f * dA.y;  st.s2 += 0.1f * dA.z;
        float u0 = st.q0 + 0.1f * dA.w;
        float u1 = st.q1 + 0.1f * dB.x;
        float u2 = st.q2 + 0.1f * dB.y;
        float u3 = st.q3 + 0.1f * dB.z;
        float inv = __frsqrt_rn(u0*u0 + u1*u1 + u2*u2 + u3*u3);
        st.q0 = u0 * inv; st.q1 = u1 * inv; st.q2 = u2 * inv; st.q3 = u3 * inv;
    }
}

// 32 async b128 scatter-stores: lanes 0..29 cover one batch's contiguous
// 480B chunk per instruction. Global chunk base dword % 4 == 0 (t0 = 8*ch),
// LDS base % 16 == 0, per-bb increments 30720B / 496B -> alignment holds.
__device__ __forceinline__ void store_chunk(const float* obuf, int b0, int t0,
                                            int tid, float* __restrict__ out) {
    // DS stores must be committed to LDS before the store-DMA engine reads them.
    asm volatile("s_wait_dscnt 0x0" ::: "memory");
    if (tid < 30) {
        const float* g = out + ((size_t)b0 * SQ + t0) * SDQ + 4 * tid;
        unsigned     l = lds_off(obuf) + 16u * tid;
        #pragma unroll 4
        for (int bb = 0; bb < NB; ++bb) {
            asm volatile("global_store_async_from_lds_b128 %0, %1, off"
                         :: "v"(g), "v"(l) : "memory");
            g += (size_t)SQ * SDQ;   // next batch, +30720B
            l += OSTRIDE * 4;        // next LDS region, +496B
        }
    }
}

__global__ __launch_bounds__(NB)
void dba_scan_kernel(const float* __restrict__ dba,   // [B,S,32]
                     const float* __restrict__ gt,    // [B,S,15]
                     float* __restrict__ out)         // [B,S,15]
{
    __shared__ __align__(16) float otile[2][OBUFSZ];  // 2 x 15.5 KB

    const int tid = threadIdx.x;            // 0..31, one batch per lane
    const int b0  = blockIdx.x * NB;
    const int b   = b0 + tid;

    // Channels 7..14 of every row are constant zero: write them ONCE per buffer.
    #pragma unroll
    for (int bf = 0; bf < 2; ++bf)
        #pragma unroll
        for (int t = 0; t < TC; ++t)
            #pragma unroll
            for (int c = 7; c < SDQ; ++c)
                otile[bf][tid * OSTRIDE + t * SDQ + c] = 0.0f;

    // init state = gt_state[b, 0, 0:7]
    const float* gp = gt + (size_t)b * SQ * SDQ;
    State st = { gp[0], gp[1], gp[2], gp[3], gp[4], gp[5], gp[6] };

    const float* ibase = dba + (size_t)b * SQ * PQ;   // this lane's batch
    f4 A[16], Bv[16];
    load_chunk(ibase, 0, A);

    float* orow0 = &otile[0][tid * OSTRIDE];
    float* orow1 = &otile[1][tid * OSTRIDE];

    for (int ch = 0; ch < NCHUNK; ch += 2) {
        // ---- even chunk: consume A, prefetch B --------------------------
        load_chunk(ibase, (ch + 1) * TC, Bv);
        // Stores complete in issue order on ASYNCcnt: <=32 outstanding proves
        // the stores that read otile[0] two chunks ago have fully drained.
        asm volatile("s_wait_asynccnt 0x20" ::: "memory");
        compute_emit(st, A, orow0);
        store_chunk(&otile[0][0], b0, ch * TC, tid, out);

        // ---- odd chunk: consume B, prefetch A ---------------------------
        if (ch + 2 < NCHUNK) load_chunk(ibase, (ch + 2) * TC, A);
        asm volatile("s_wait_asynccnt 0x20" ::: "memory");
        compute_emit(st, Bv, orow1);
        store_chunk(&otile[1][0], b0, (ch + 1) * TC, tid, out);
    }

    // Drain all outstanding store DMA before the wave exits.
    asm volatile("s_wait_asynccnt 0x0" ::: "memory");
}

extern "C" void kernel_launch(void* const* d_in, const int* in_sizes, int n_in,
                              void* d_out, int out_size, void* d_ws, size_t ws_size,
                              hipStream_t stream) {
    const float* dba = (const float*)d_in[0];   // dba_params   [B,S,32]
    // d_in[1] = imu_measurements (unused by the reference math)
    const float* gt  = (const float*)d_in[2];   // gt_state     [B,S,15]
    float* out = (float*)d_out;                 // [B,S,15]

    dim3 grid(BQ / NB);   // 128 workgroups
    dim3 block(NB);       // one wave32 each
    dba_scan_kernel<<<grid, block, 0, stream>>>(dba, gt, out);
}